// MXFP4Quantizer_88175678587514
// MI455X (gfx1250) — compile-verified
//
#include <hip/hip_runtime.h>
#include <hip/hip_bf16.h>

// ---------------------------------------------------------------------------
// MXFP4 group quantizer for MI455X (gfx1250).
// weight: 4096 x 16384 fp32. GROUP_SIZE=32 along columns -> G=512 groups,
// max_abs reduced over ALL rows per group. Two streaming passes over 268MB
// (HBM-bound, ~35us at 23.3 TB/s). wave32: 256-thread blocks = 8 waves.
// ---------------------------------------------------------------------------

#define R_DIM 4096
#define L_DIM 16384
#define N_ELEM (R_DIM * L_DIM)        // 2^26 elements
#define N4 (N_ELEM / 4)               // 2^24 float4
#define COL4_PER_ROW (L_DIM / 4)      // 4096 float4 per row
#define G_DIM 512
#define ROW_SLICES 64                 // absmax: 64-way row parallelism

typedef float v4f __attribute__((ext_vector_type(4)));
typedef int   v4i __attribute__((ext_vector_type(4)));

// Pointer types matching the gfx1250 async-copy builtin signature
// (param 1: AS1 int4*, param 2: AS3 int4* -- confirmed by round-1 diagnostic).
typedef __attribute__((address_space(1))) v4i* gas_v4i_ptr;
typedef __attribute__((address_space(3))) v4i* lds_v4i_ptr;

// gfx1250 async copy to LDS (guarded: falls back to plain loads if the
// toolchain doesn't declare the builtins; host pass always takes fallback).
#if defined(__has_builtin)
#if __has_builtin(__builtin_amdgcn_global_load_async_to_lds_b128) && \
    __has_builtin(__builtin_amdgcn_s_wait_asynccnt)
#define MX_USE_ASYNC_LDS 1
#endif
#endif

__device__ __forceinline__ int imax(int a, int b) { return a > b ? a : b; }

// ---------------------------------------------------------------------------
// Pass 0: zero the 512 per-group atomic-max slots in workspace.
// ---------------------------------------------------------------------------
__global__ void mx_init(int* __restrict__ gmax) {
    gmax[threadIdx.x] = 0;
}

// ---------------------------------------------------------------------------
// Pass A: per-group max|w|. Each thread owns one fixed float4 column chunk
// (perfectly coalesced: a wave reads 512 contiguous bytes per iteration) and
// strides over rows (stride = 64 rows = 4MB address stride -> SW prefetch).
// abs-float bit pattern preserves ordering, so integer max works.
// Lanes 8k..8k+7 share one group -> shfl_xor tree, then 1 atomic per 8 lanes.
// ---------------------------------------------------------------------------
__global__ __launch_bounds__(256) void mx_absmax(const v4f* __restrict__ W,
                                                 int* __restrict__ gmax) {
    const int tid   = blockIdx.x * 256 + threadIdx.x;     // 0 .. 262143
    const int col4  = tid & (COL4_PER_ROW - 1);           // fixed column chunk
    const int slice = tid >> 12;                          // 0 .. 63
    const v4f* p = W + (size_t)slice * COL4_PER_ROW + col4;
    const size_t step = (size_t)ROW_SLICES * COL4_PER_ROW;

    int m = 0;
    for (int r = slice; r < R_DIM; r += ROW_SLICES) {
        __builtin_prefetch(p + step, 0, 0);               // global_prefetch_b8
        v4f w = *p;
        int bx = (int)(__float_as_uint(w.x) & 0x7fffffffu);
        int by = (int)(__float_as_uint(w.y) & 0x7fffffffu);
        int bz = (int)(__float_as_uint(w.z) & 0x7fffffffu);
        int bw = (int)(__float_as_uint(w.w) & 0x7fffffffu);
        m = imax(m, imax(imax(bx, by), imax(bz, bw)));
        p += step;
    }
    // reduce across the 8 lanes that share group (col4 >> 3)
    m = imax(m, __shfl_xor(m, 1, 8));
    m = imax(m, __shfl_xor(m, 2, 8));
    m = imax(m, __shfl_xor(m, 4, 8));
    if ((threadIdx.x & 7) == 0) {
        atomicMax(&gmax[col4 >> 3], m);
    }
}

// ---------------------------------------------------------------------------
// Pass B-prep: one block, 512 threads. eps_eff, e_base -> output tail;
// scale / 1/scale / eps_eff tables -> workspace (contiguous 1536 floats).
// frexpf gives exact floor(log2(x)) for all finite x>0 incl. powers of two.
// ---------------------------------------------------------------------------
__global__ void mx_finalize(const int* __restrict__ gmax,
                            const float* __restrict__ eps_param,
                            const float* __restrict__ adv_delta,
                            float* __restrict__ out_tail,   // [0,512): eps_eff, [512,1024): e_base
                            float* __restrict__ tables) {   // [0,512): s, [512,1024): 1/s, [1024,1536): eps
    const int g = threadIdx.x;
    const float ma = __int_as_float(gmax[g]);               // non-negative
    float e_base = 0.0f;
    if (ma > 0.0f) {
        int e;
        (void)frexpf(ma, &e);        // ma = f * 2^e, f in [0.5, 1)
        e_base = (float)(e - 1);     // == floor(log2(ma)) exactly
    }
    const float s  = ldexpf(1.0f, (int)e_base);             // exact power of 2
    const float ep = 0.5f * tanhf(eps_param[g]);
    const float ee = fminf(fmaxf(ep + adv_delta[g], -0.5f), 0.5f);

    out_tail[g]           = ee;      // eps_eff (output 2)
    out_tail[G_DIM + g]   = e_base;  // e_base  (output 3)
    tables[g]             = s;
    tables[G_DIM + g]     = 1.0f / s; // exact reciprocal of power of 2
    tables[2 * G_DIM + g] = ee;
}

// ---------------------------------------------------------------------------
// Pass B: quantize. Grid-stride over 2^24 float4 (all 4 lanes of a float4
// share one group since 32 | group width). Tables staged into LDS via gfx1250
// async-copy when available. NT loads/stores: the data is last-use.
// ---------------------------------------------------------------------------
__device__ __forceinline__ float mx_q1(float w, float s, float inv, float e) {
    const float a     = fabsf(w);
    const float ratio = fminf(a * inv, 1.0f);               // lower clip free: a>=0
    const float sh    = fminf(fmaxf(ratio + e, 0.0f), 1.0f);
    const float mq    = rintf(sh * 8.0f) * 0.125f;          // round-half-even
    const float q     = copysignf(s * mq, w);
    return (w == 0.0f) ? 0.0f : q;                          // jnp.sign(0) == 0
}

__global__ __launch_bounds__(256) void mx_quant(const v4f* __restrict__ W,
                                                v4f* __restrict__ Q,
                                                const v4f* __restrict__ tab4) {
    __shared__ float s_tab[3 * G_DIM];                      // 6KB
#if defined(MX_USE_ASYNC_LDS)
    if (threadIdx.x < 384) {                                // 384 float4 = 1536 floats
        v4i* src = (v4i*)(tab4 + threadIdx.x);              // strip const, repoint
        v4i* dst = (v4i*)(s_tab) + threadIdx.x;
        __builtin_amdgcn_global_load_async_to_lds_b128(
            (gas_v4i_ptr)src, (lds_v4i_ptr)dst, 0, 0);
    }
    __builtin_amdgcn_s_wait_asynccnt(0);
    __syncthreads();
#else
    {
        const float* tf = (const float*)tab4;
        for (int i = threadIdx.x; i < 3 * G_DIM; i += 256) s_tab[i] = tf[i];
        __syncthreads();
    }
#endif
    const float* __restrict__ s_s = s_tab;
    const float* __restrict__ s_i = s_tab + G_DIM;
    const float* __restrict__ s_e = s_tab + 2 * G_DIM;

    const int stride = (int)(gridDim.x * 256u);
    for (int i = blockIdx.x * 256 + threadIdx.x; i < N4; i += stride) {
        v4f w = __builtin_nontemporal_load(W + i);
        const int g = (i & (COL4_PER_ROW - 1)) >> 3;        // group of this float4
        const float s = s_s[g], inv = s_i[g], e = s_e[g];
        v4f q;
        q.x = mx_q1(w.x, s, inv, e);
        q.y = mx_q1(w.y, s, inv, e);
        q.z = mx_q1(w.z, s, inv, e);
        q.w = mx_q1(w.w, s, inv, e);
        __builtin_nontemporal_store(q, Q + i);
    }
}

// ---------------------------------------------------------------------------
extern "C" void kernel_launch(void* const* d_in, const int* in_sizes, int n_in,
                              void* d_out, int out_size, void* d_ws, size_t ws_size,
                              hipStream_t stream) {
    const float* W         = (const float*)d_in[0];   // 4096*16384
    const float* eps_param = (const float*)d_in[1];   // 512
    const float* adv_delta = (const float*)d_in[2];   // 512
    float* out = (float*)d_out;                       // [N) quant, [N,N+512) eps_eff, [N+512,N+1024) e_base

    // workspace layout (4-byte units): [0,512) int gmax; [512,2048) float tables
    int*   gmax   = (int*)d_ws;
    float* tables = (float*)d_ws + G_DIM;

    mx_init<<<1, G_DIM, 0, stream>>>(gmax);

    // 1024 blocks * 256 threads = 4096 col-chunks * 64 row slices
    mx_absmax<<<(COL4_PER_ROW * ROW_SLICES) / 256, 256, 0, stream>>>(
        (const v4f*)W, gmax);

    mx_finalize<<<1, G_DIM, 0, stream>>>(gmax, eps_param, adv_delta,
                                         out + (size_t)N_ELEM, tables);

    mx_quant<<<8192, 256, 0, stream>>>((const v4f*)W, (v4f*)out,
                                       (const v4f*)tables);
}